// Distance_61942018343617
// MI455X (gfx1250) — compile-verified
//
#include <hip/hip_runtime.h>
#include <hip/hip_bf16.h>
#include <math.h>

typedef __attribute__((ext_vector_type(2))) float v2f;
typedef __attribute__((ext_vector_type(8))) float v8f;

#define N_BATCH 8
#define M_PTS   3584
#define K_NBR   16
#define CUT2    25.0f
#define BIGV    1e30f

#define WAVES_PER_BLK 2
#define THREADS       (WAVES_PER_BLK * 32)
#define ROWS_PER_BLK  (WAVES_PER_BLK * 32)          // 64 centers per block

// Two waves per block. Whole batch position panel (43KB) is DMA'd into LDS once
// via CDNA5 async global->LDS copies; the 224-tile hot loop is then pure
// LDS + VALU + v_wmma_f32_16x16x4_f32 (Gram tiles), no global traffic.
__global__ __launch_bounds__(THREADS) void radius_graph_wmma_kernel(
    const float* __restrict__ pos,   // [N, M, 3] f32
    const int*   __restrict__ mask,  // [N, M]
    float*       __restrict__ out)   // [7*E] f32: ei0|ei1|w|vec*3|em
{
  const int tid  = threadIdx.x;
  const int wave = tid >> 5;
  const int lane = tid & 31;
  const int blocksPerBatch = M_PTS / ROWS_PER_BLK;   // 56
  const int n      = blockIdx.x / blocksPerBatch;
  const int rblk   = blockIdx.x % blocksPerBatch;
  const int rbase  = rblk * ROWS_PER_BLK + wave * 32; // this wave's 32 centers

  const float* posB  = pos  + (size_t)n * M_PTS * 3;
  const int*   maskB = mask + (size_t)n * M_PTS;

  __shared__ float sPos[M_PTS * 3];                  // 43008 B, whole batch panel
  __shared__ float sN2[M_PTS];                       // 14336 B, |p|^2 or BIG if invalid
  __shared__ float sD2[WAVES_PER_BLK][32][17];       // 4352 B, per-wave scratch
  __shared__ float sRowN2[ROWS_PER_BLK];             // 256 B

  // ---- async DMA: global position panel -> LDS (b128 chunks, GVS addressing) ----
  {
    const unsigned ldsPosBase = (unsigned)(size_t)&sPos[0];
    const int nChunks = (M_PTS * 3) / 4;             // 2688 x 16B
    for (int i = tid; i < nChunks; i += THREADS) {
      const unsigned boff = (unsigned)i * 16u;
      const unsigned ldst = ldsPosBase + boff;
      asm volatile("global_load_async_to_lds_b128 %0, %1, %2"
                   :: "v"(ldst), "v"(boff), "s"(posB) : "memory");
    }
    asm volatile("s_wait_asynccnt 0x0" ::: "memory");
  }
  __syncthreads();

  // ---- norms with validity folded in (invalid endpoint => d2 fails cutoff) ----
  for (int j = tid; j < M_PTS; j += THREADS) {
    const float x = sPos[3 * j + 0];
    const float y = sPos[3 * j + 1];
    const float z = sPos[3 * j + 2];
    sN2[j] = maskB[j] ? (x * x + y * y + z * z) : BIGV;
  }
  __syncthreads();

  // ---- per-lane center + A fragments (16x4 f32: lane=M, lane-half+VGPR=K) ----
  const int myRow = rbase + lane;
  const float pix = sPos[myRow * 3 + 0];
  const float piy = sPos[myRow * 3 + 1];
  const float piz = sPos[myRow * 3 + 2];
  sRowN2[wave * 32 + lane] = sN2[myRow];
  __syncthreads();

  const int hf  = lane >> 4;
  const int sub = lane & 15;
  v2f afrag[2];
  float rowN2v[2][8];
#pragma unroll
  for (int t = 0; t < 2; ++t) {
    const int r = rbase + t * 16 + sub;
    afrag[t].x = sPos[r * 3 + 2 * hf];               // K=0 -> x, K=2 -> z
    afrag[t].y = (hf == 0) ? sPos[r * 3 + 1] : 0.0f; // K=1 -> y, K=3 -> pad
#pragma unroll
    for (int v = 0; v < 8; ++v)                      // hoisted row norms (loop-invariant)
      rowN2v[t][v] = sRowN2[wave * 32 + t * 16 + v + 8 * hf];
  }

  float bd[K_NBR];
  int   bi[K_NBR];
#pragma unroll
  for (int k = 0; k < K_NBR; ++k) { bd[k] = BIGV; bi[k] = 0; }

  const v8f czero = {0.f, 0.f, 0.f, 0.f, 0.f, 0.f, 0.f, 0.f};

  // ---- hot loop: all data in LDS, no barriers (sD2 slice is wave-private) ----
  for (int jt = 0; jt < M_PTS / 16; ++jt) {
    const int jbase = jt * 16;
    const int j     = jbase + sub;

    v2f bfrag;                                       // 4x16 f32: lane=N
    bfrag.x = sPos[3 * j + 2 * hf];
    bfrag.y = (hf == 0) ? sPos[3 * j + 1] : 0.0f;
    const float cn2 = sN2[j];

#pragma unroll
    for (int t = 0; t < 2; ++t) {
      v8f d = __builtin_amdgcn_wmma_f32_16x16x4_f32(
          false, afrag[t], false, bfrag, (short)0, czero, false, false);
#pragma unroll
      for (int v = 0; v < 8; ++v) {
        const int   rIn = t * 16 + v + 8 * hf;       // C/D layout: M = v + 8*half
        const float d2  = fmaxf(rowN2v[t][v] + cn2 - 2.0f * d[v], 0.0f);
        const bool  ok  = (d2 <= CUT2) && ((rbase + rIn) != j);
        sD2[wave][rIn][sub] = ok ? d2 : BIGV;
      }
    }
    __builtin_amdgcn_wave_barrier();                 // order cross-lane LDS traffic

#pragma unroll 1
    for (int c = 0; c < 16; ++c) {
      const float cd = sD2[wave][lane][c];
      if (cd < bd[K_NBR - 1]) {                      // fast reject vs current 16th-best
        float dcur = cd;
        int   icur = jbase + c;
#pragma unroll
        for (int k = 0; k < K_NBR; ++k) {            // branch-free stable sorted insert
          const bool  sm = dcur < bd[k];
          const float td = bd[k]; const int ti = bi[k];
          bd[k] = sm ? dcur : td;  bi[k] = sm ? icur : ti;
          dcur  = sm ? td : dcur;  icur  = sm ? ti : icur;
        }
      }
    }
    __builtin_amdgcn_wave_barrier();
  }

  // ---- emit outputs; exact vec/weight recomputed from LDS positions ----
  const long E = (long)N_BATCH * M_PTS * K_NBR;      // 458752
  float* ei0 = out;
  float* ei1 = out + E;
  float* ew  = out + 2 * E;
  float* ev  = out + 3 * E;
  float* emf = out + 6 * E;

  const long base = ((long)n * M_PTS + myRow) * K_NBR;
#pragma unroll 1
  for (int k = 0; k < K_NBR; ++k) {
    const bool real = bd[k] < 0.5f * BIGV;
    const int  jj   = bi[k];
    float vx = 0.f, vy = 0.f, vz = 0.f, w = 0.f;
    float s0 = 0.f, s1 = 0.f;
    if (real) {
      vx = sPos[jj * 3 + 0] - pix;
      vy = sPos[jj * 3 + 1] - piy;
      vz = sPos[jj * 3 + 2] - piz;
      const float dd = vx * vx + vy * vy + vz * vz;
      w  = (dd > 0.f) ? sqrtf(dd) : 0.f;
      s0 = (float)(n * M_PTS + jj);
      s1 = (float)(n * M_PTS + myRow);
    }
    const long e = base + k;
    ei0[e] = s0;
    ei1[e] = s1;
    ew[e]  = w;
    ev[e * 3 + 0] = vx;
    ev[e * 3 + 1] = vy;
    ev[e * 3 + 2] = vz;
    emf[e] = real ? 1.0f : 0.0f;
  }
}

extern "C" void kernel_launch(void* const* d_in, const int* in_sizes, int n_in,
                              void* d_out, int out_size, void* d_ws, size_t ws_size,
                              hipStream_t stream) {
  const float* pos  = (const float*)d_in[0];   // [8,256,14,3] f32
  const int*   mask = (const int*)d_in[1];     // [8,256,14]
  float* out = (float*)d_out;                  // 7*E floats
  (void)in_sizes; (void)n_in; (void)out_size; (void)d_ws; (void)ws_size;

  const int grid = N_BATCH * (M_PTS / ROWS_PER_BLK);  // 448 blocks x 64 threads
  radius_graph_wmma_kernel<<<grid, THREADS, 0, stream>>>(pos, mask, out);
}